// fusion_loss_61220463837559
// MI455X (gfx1250) — compile-verified
//
#include <hip/hip_runtime.h>
#include <hip/hip_bf16.h>
#include <stdint.h>

// Detection-matching loss:
//   preds:   (1, 32768, 7) f32  [x1,y1,x2,y2,conf,conf,cls]
//   targets: (1024, 7)     f32
//   out:     scalar f32
//
// VALU-bound (~0.5 GFLOP f32, ~1 MB HBM). No MAC structure -> WMMA N/A.
// CDNA5 paths used: async global->LDS staging with on-the-fly repack to a
// padded stride-8 layout (global_load_async_to_lds_b32 + s_wait_asynccnt),
// so the hot loop reads each target with one ds_load_b128 + one ds_load_b64.

#define N_TGT_MAX 1024
#define THREADS   256

__global__ __launch_bounds__(THREADS)
void det_loss_kernel(const float* __restrict__ preds,
                     const float* __restrict__ targets,
                     float* __restrict__ partial,
                     int n_pred, int n_tgt) {
    // Padded target rows, 8 floats (32 B) each:
    //   [0..3] = x1,y1,x2,y2   [4] = cls   [5] = area   [6,7] = scrap (conf dup)
    __shared__ __align__(16) float s_tgt8[8 * N_TGT_MAX];
    __shared__ float s_red[THREADS];

    const int tid = threadIdx.x;
    const int nwords = n_tgt * 7;

    // ---- Async global->LDS staging with repack (7-float AoS -> 8-float padded) ----
    {
        const uint64_t saddr = (uint64_t)(uintptr_t)targets;
        // Low 32 bits of a shared-aperture generic address == LDS byte offset.
        const uint32_t lds_base = (uint32_t)(uintptr_t)(&s_tgt8[0]);
        for (int k = tid; k < nwords; k += THREADS) {
            const int j = k / 7;
            const int c = k - j * 7;
            // src col -> dst slot: 0..3 -> 0..3, 6(cls) -> 4, 4 -> 6, 5 -> 7
            const int slot = (c < 4) ? c : ((c == 6) ? 4 : (c == 4 ? 6 : 7));
            const uint32_t voff  = (uint32_t)k * 4u;
            const uint32_t laddr = lds_base + (uint32_t)(j * 32 + slot * 4);
            asm volatile("global_load_async_to_lds_b32 %0, %1, %2"
                         :
                         : "v"(laddr), "v"(voff), "s"(saddr)
                         : "memory");
        }
        // Drain this wave's ASYNCcnt, then barrier so every wave sees the data.
        asm volatile("s_wait_asynccnt 0x0" ::: "memory");
    }
    __syncthreads();

    // ---- Precompute target areas into slot 5 ----
    for (int j = tid; j < n_tgt; j += THREADS) {
        const float tx1 = s_tgt8[j * 8 + 0], ty1 = s_tgt8[j * 8 + 1];
        const float tx2 = s_tgt8[j * 8 + 2], ty2 = s_tgt8[j * 8 + 3];
        s_tgt8[j * 8 + 5] = (tx2 - tx1 + 1.0f) * (ty2 - ty1 + 1.0f);
    }
    __syncthreads();

    // ---- One pred per lane; scan all targets for the first match ----
    const int i = blockIdx.x * THREADS + tid;
    float loss = 0.0f;
    if (i < n_pred) {
        const float* p = preds + (size_t)i * 7;
        const float px1 = p[0], py1 = p[1], px2 = p[2], py2 = p[3];
        const float conf = p[4];
        const float pcls = p[6];
        const float areaP = (px2 - px1 + 1.0f) * (py2 - py1 + 1.0f);

        int first = -1;
        #pragma unroll 4
        for (int j = 0; j < n_tgt; ++j) {
            // Uniform-j LDS reads: wave32 broadcast; b128 + b64 per target.
            const float4 tb = *(const float4*)(s_tgt8 + j * 8);      // box
            const float2 ca = *(const float2*)(s_tgt8 + j * 8 + 4);  // (cls, area)

            const float xA = fmaxf(px1, tb.x);
            const float yA = fmaxf(py1, tb.y);
            const float xB = fminf(px2, tb.z);
            const float yB = fminf(py2, tb.w);
            const float iw = fmaxf(0.0f, xB - xA + 1.0f);
            const float ih = fmaxf(0.0f, yB - yA + 1.0f);
            const float inter = iw * ih;
            const float uni = areaP + ca.y - inter;
            // iou > 0.5  <=>  inter > 0.5*union  (union > 0 always here)
            const bool m = (inter > 0.5f * uni) && (pcls == ca.x);
            if (m && first < 0) first = j;
        }

        if (first >= 0) {
            const float4 tb = *(const float4*)(s_tgt8 + first * 8);
            const float d0 = fabsf(px1 - tb.x);
            const float d1 = fabsf(py1 - tb.y);
            const float d2 = fabsf(px2 - tb.z);
            const float d3 = fabsf(py2 - tb.w);
            const float h0 = (d0 < 1.0f) ? 0.5f * d0 * d0 : d0 - 0.5f;
            const float h1 = (d1 < 1.0f) ? 0.5f * d1 * d1 : d1 - 0.5f;
            const float h2 = (d2 < 1.0f) ? 0.5f * d2 * d2 : d2 - 0.5f;
            const float h3 = (d3 < 1.0f) ? 0.5f * d3 * d3 : d3 - 0.5f;
            const float sl1 = (h0 + h1 + h2 + h3) * 0.25f;
            loss = sl1 - logf(conf);
        } else {
            loss = -logf(1.0f - conf);
        }
    }

    // ---- Deterministic block reduction ----
    s_red[tid] = loss;
    __syncthreads();
    #pragma unroll
    for (int s = THREADS / 2; s > 0; s >>= 1) {
        if (tid < s) s_red[tid] += s_red[tid + s];
        __syncthreads();
    }
    if (tid == 0) partial[blockIdx.x] = s_red[0];
}

__global__ __launch_bounds__(THREADS)
void det_loss_finalize(const float* __restrict__ partial,
                       float* __restrict__ out, int nblocks) {
    __shared__ float s_red[THREADS];
    float s = 0.0f;
    for (int i = threadIdx.x; i < nblocks; i += THREADS) s += partial[i];
    s_red[threadIdx.x] = s;
    __syncthreads();
    #pragma unroll
    for (int k = THREADS / 2; k > 0; k >>= 1) {
        if (threadIdx.x < k) s_red[threadIdx.x] += s_red[threadIdx.x + k];
        __syncthreads();
    }
    if (threadIdx.x == 0) out[0] = s_red[0];
}

extern "C" void kernel_launch(void* const* d_in, const int* in_sizes, int n_in,
                              void* d_out, int out_size, void* d_ws, size_t ws_size,
                              hipStream_t stream) {
    const float* preds   = (const float*)d_in[0];   // (1, n_pred, 7) f32
    const float* targets = (const float*)d_in[1];   // (n_tgt, 7) f32
    float* out     = (float*)d_out;
    float* partial = (float*)d_ws;

    int n_pred = in_sizes[0] / 7;
    int n_tgt  = in_sizes[1] / 7;
    if (n_tgt > N_TGT_MAX) n_tgt = N_TGT_MAX;

    const int nblocks = (n_pred + THREADS - 1) / THREADS;   // 128 for 32768 preds

    det_loss_kernel<<<nblocks, THREADS, 0, stream>>>(preds, targets, partial,
                                                     n_pred, n_tgt);
    det_loss_finalize<<<1, THREADS, 0, stream>>>(partial, out, nblocks);
}